// TameHighOrderActivationA_85220741087984
// MI455X (gfx1250) — compile-verified
//
#include <hip/hip_runtime.h>

// Problem constants (match reference)
#define BATCH   4096
#define GROUPS  512
#define ARITY   4
#define OUT_DIM 16
#define XCOLS   (GROUPS * ARITY)    // 2048
#define OCOLS   (GROUPS * OUT_DIM)  // 8192

typedef __attribute__((ext_vector_type(2))) float v2f;
typedef __attribute__((ext_vector_type(4))) float v4f;
typedef __attribute__((ext_vector_type(8))) float v8f;

__device__ __forceinline__ void csort(float& a, int& ia, float& b, int& ib) {
    // stable compare-swap (strict > keeps argsort stability for ties)
    if (a > b) { float t = a; a = b; b = t; int ti = ia; ia = ib; ib = ti; }
}

// One wave = 16 batch rows x 1 group. D = C(16x16 dense coef) x params[g](16x16),
// accumulated as four V_WMMA_F32_16X16X4_F32 (K slices of 4, fp32 exact).
__global__ __launch_bounds__(256) void hoact_wmma_kernel(const float* __restrict__ X,
                                                         const float* __restrict__ P,
                                                         float* __restrict__ O) {
    const int lane = threadIdx.x & 31;
    const int wave = threadIdx.x >> 5;
    const int g    = blockIdx.x >> 5;                   // 0..511 (8 waves share group -> params L1 reuse)
    const int tile = ((blockIdx.x & 31) << 3) + wave;   // 0..255 batch tile
    const int b0   = tile << 4;                         // base batch row of 16-row tile
    const int half = lane >> 4;                         // 0: A holds K{0,1}; 1: A holds K{2,3}
    const int n    = lane & 15;                         // row within tile / output column

    // ---- per-lane preprocessing: sort 4 inputs of row (b0+n, group g) ----
    v4f x = *(const v4f*)(X + (size_t)(b0 + n) * XCOLS + (size_t)g * ARITY);
    float a0 = x.x, a1 = x.y, a2 = x.z, a3 = x.w;
    int   i0 = 0,  i1 = 1,  i2 = 2,  i3 = 3;
    csort(a0, i0, a1, i1);
    csort(a2, i2, a3, i3);
    csort(a0, i0, a2, i2);
    csort(a1, i1, a3, i3);
    csort(a1, i1, a2, i2);
    const float c0 = a0, c1 = a1 - a0, c2 = a2 - a1, c3 = a3 - a2;
    const int p1 = 1 << i1, p2 = 1 << i2, p3 = 1 << i3;
    const int vid3 = p3;          // in {1,2,4,8}
    const int vid2 = p2 + p3;     // in {3,5,6,9,10,12}
    const int vid1 = p1 + vid2;   // in {7,11,13,14}; vid0 == 15 always

    // dense 16-wide coefficient row (no dynamic register indexing: unrolled selects)
    float Cd[16];
#pragma unroll
    for (int v = 0; v < 16; ++v) {
        float t = 0.0f;
        t = (vid1 == v) ? c1 : t;
        t = (vid2 == v) ? c2 : t;
        t = (vid3 == v) ? c3 : t;
        Cd[v] = (v == 15) ? (t + c0) : t;
    }

    // ---- GEMM: 4 x V_WMMA_F32_16X16X4_F32, K = vertex dimension ----
    // A (16x4 f32) layout per ISA: VGPR0 = K0(lanes0-15)/K2(lanes16-31), VGPR1 = K1/K3.
    // B (4x16 f32) mirrored: VGPR0 = row K0 (lanes0-15, N=lane) / row K2 (lanes16-31).
    const float* Pg = P + (size_t)g * (16 * OUT_DIM);
    v8f acc = {0.f, 0.f, 0.f, 0.f, 0.f, 0.f, 0.f, 0.f};
#pragma unroll
    for (int k = 0; k < 4; ++k) {
        v2f a;
        a.x = half ? Cd[4 * k + 2] : Cd[4 * k + 0];
        a.y = half ? Cd[4 * k + 3] : Cd[4 * k + 1];
        const float* pb = Pg + (size_t)(4 * k + 2 * half) * OUT_DIM + n;
        v2f b;
        b.x = pb[0];        // row 4k + 2*half
        b.y = pb[OUT_DIM];  // row 4k + 2*half + 1
        acc = __builtin_amdgcn_wmma_f32_16x16x4_f32(
            /*neg_a=*/false, a, /*neg_b=*/false, b,
            /*c_mod=*/(short)0, acc, /*reuse_a=*/false, /*reuse_b=*/false);
    }

    // ---- store D: VGPR j -> M=j (lanes0-15) / M=j+8 (lanes16-31), N=lane&15 ----
    float* obase = O + (size_t)b0 * OCOLS + (size_t)g * OUT_DIM + n;
#pragma unroll
    for (int j = 0; j < 8; ++j) {
        const int M = j + 8 * half;
        obase[(size_t)M * OCOLS] = acc[j];
    }
}

extern "C" void kernel_launch(void* const* d_in, const int* in_sizes, int n_in,
                              void* d_out, int out_size, void* d_ws, size_t ws_size,
                              hipStream_t stream) {
    const float* X = (const float*)d_in[0];  // (4096, 2048) f32
    const float* P = (const float*)d_in[1];  // (512, 16, 16) f32
    float* O = (float*)d_out;                // (4096, 8192) f32

    // 512 groups * 32 batch-chunks; 8 waves/block (one tile each)
    dim3 grid(GROUPS * 32);
    dim3 block(256);
    hoact_wmma_kernel<<<grid, block, 0, stream>>>(X, P, O);
}